// NTXentLoss_56418690400495
// MI455X (gfx1250) — compile-verified
//
#include <hip/hip_runtime.h>
#include <math.h>

#define BB 4096
#define NN 8192
#define DD 512
#define INV_T 2.0f        // 1 / temperature(0.5)
#define NEG_INIT -1.0e30f // finite sentinel for running max (logits are in [-2,2])

typedef __attribute__((ext_vector_type(2))) float v2f;
typedef __attribute__((ext_vector_type(8))) float v8f;

// ---------------------------------------------------------------------------
// Kernel 1: per-row inverse norms (clamped at 1e-8), one wave32 per row.
// ---------------------------------------------------------------------------
__global__ __launch_bounds__(256) void row_invnorm_kernel(
    const float* __restrict__ za, const float* __restrict__ zp,
    float* __restrict__ invnorm)
{
    const int w    = threadIdx.x >> 5;
    const int lane = threadIdx.x & 31;
    const int row  = blockIdx.x * 8 + w;
    const float* src = (row < BB) ? (za + (size_t)row * DD)
                                  : (zp + (size_t)(row - BB) * DD);
    float ss = 0.f;
    for (int i = lane; i < DD; i += 32) {
        float v = src[i];
        ss += v * v;
    }
#pragma unroll
    for (int mask = 16; mask >= 1; mask >>= 1)
        ss += __shfl_xor(ss, mask, 32);
    if (lane == 0) {
        float nrm = fmaxf(sqrtf(ss), 1e-8f);
        invnorm[row] = 1.0f / nrm;
    }
}

// ---------------------------------------------------------------------------
// Kernel 2: per-column sum / sum-of-squares in fp64 (for z_std, z_mean_norm).
// ---------------------------------------------------------------------------
__global__ __launch_bounds__(256) void col_stats_kernel(
    const float* __restrict__ za, const float* __restrict__ zp,
    double* __restrict__ colsum, double* __restrict__ colsq)
{
    __shared__ double s1[256];
    __shared__ double s2[256];
    const int d = blockIdx.x;
    const int t = threadIdx.x;
    double a = 0.0, b = 0.0;
    for (int r = t; r < NN; r += 256) {
        const float* src = (r < BB) ? (za + (size_t)r * DD)
                                    : (zp + (size_t)(r - BB) * DD);
        double v = (double)src[d];
        a += v;
        b += v * v;
    }
    s1[t] = a; s2[t] = b;
    __syncthreads();
    for (int off = 128; off > 0; off >>= 1) {
        if (t < off) { s1[t] += s1[t + off]; s2[t] += s2[t + off]; }
        __syncthreads();
    }
    if (t == 0) { colsum[d] = s1[0]; colsq[d] = s2[0]; }
}

// ---------------------------------------------------------------------------
// Kernel 3: fused cosine-GEMM (fp32 WMMA 16x16x4) + online logsumexp/argmax.
// Block = 256 threads (8 waves) owns 16 rows; wave w streams N-tiles
// nt = w, w+8, ...  The per-column invnorm scale is factored OUT of the
// K-loop (applied once to the accumulator), so the hot loop is just
// ds_load(A) + global_load(B) + v_wmma.  Consume phase is branchless.
// ---------------------------------------------------------------------------
__global__ __launch_bounds__(256) void ntxent_main_kernel(
    const float* __restrict__ za, const float* __restrict__ zp,
    const float* __restrict__ invnorm,
    float* __restrict__ loss_term, float* __restrict__ correct,
    float* __restrict__ cs_row, float* __restrict__ pc_row)
{
    __shared__ float Atile[16 * 514];
    __shared__ float mbuf[8][16];
    __shared__ float sbuf[8][16];
    __shared__ float csbuf[8][16];
    __shared__ float pcbuf[8][16];
    __shared__ int   ambuf[8][16];

    const int rowBase = blockIdx.x * 16;
    const int tid = threadIdx.x;

    // Stage normalized A-tile (16 x 512) into LDS, padded stride 514.
    for (int idx = tid; idx < 16 * DD; idx += 256) {
        int r  = idx >> 9;          // / 512
        int c  = idx & (DD - 1);
        int gr = rowBase + r;
        const float* src = (gr < BB) ? (za + (size_t)gr * DD)
                                     : (zp + (size_t)(gr - BB) * DD);
        Atile[r * 514 + c] = src[c] * invnorm[gr];
    }
    __syncthreads();

    const int w    = tid >> 5;
    const int lane = tid & 31;
    const int lr   = lane & 15;
    const int half = lane >> 4;
    const int kk   = half * 2;      // K sub-offset for this half of the wave

    // Per-lane online stats for 8 rows (rows rowBase + j + 8*half).
    float m[8], s[8], cs[8], pc[8];
    int   am[8];
#pragma unroll
    for (int j = 0; j < 8; ++j) {
        m[j] = NEG_INIT; s[j] = 0.f; cs[j] = 0.f; pc[j] = 0.f; am[j] = -1;
    }

    const float* arow = &Atile[lr * 514 + kk];

    for (int nt = w; nt < NN / 16; nt += 8) {
        const int colBase = nt * 16;
        const int col = colBase + lr;
        const float* bsrc = (col < BB) ? (za + (size_t)col * DD)
                                       : (zp + (size_t)(col - BB) * DD);
        const float binv = invnorm[col];
        const float* brow = bsrc + kk;

        v8f acc = {0.f, 0.f, 0.f, 0.f, 0.f, 0.f, 0.f, 0.f};
#pragma unroll 8
        for (int k = 0; k < DD; k += 4) {
            v2f a, b;
            a.x = arow[k];
            a.y = arow[k + 1];
            b.x = brow[k];
            b.y = brow[k + 1];
            acc = __builtin_amdgcn_wmma_f32_16x16x4_f32(
                false, a, false, b, (short)0, acc, false, false);
        }

        // Branchless consume: lane holds column `col`, rows j + 8*half.
#pragma unroll
        for (int j = 0; j < 8; ++j) {
            float cv   = acc[j] * binv;         // apply column norm once
            int   grow = rowBase + j + half * 8;
            bool  diag = (col == grow);
            float logit = diag ? -INFINITY : cv * INV_T;
            cs[j] += diag ? 0.f : cv;
            pc[j] += (col == (grow ^ BB)) ? cv : 0.f;
            float Mn = fmaxf(m[j], logit);
            am[j] = (logit > m[j]) ? col : am[j];
            s[j]  = s[j] * __expf(m[j] - Mn) + __expf(logit - Mn);
            m[j]  = Mn;
        }
    }

    // Merge across the 16 lanes of each half (xor butterfly stays in-half).
#pragma unroll
    for (int mask = 1; mask <= 8; mask <<= 1) {
#pragma unroll
        for (int j = 0; j < 8; ++j) {
            float om  = __shfl_xor(m[j],  mask, 32);
            float os  = __shfl_xor(s[j],  mask, 32);
            int   oam = __shfl_xor(am[j], mask, 32);
            float ocs = __shfl_xor(cs[j], mask, 32);
            float opc = __shfl_xor(pc[j], mask, 32);
            cs[j] += ocs;
            pc[j] += opc;
            float Mn = fmaxf(m[j], om);
            float sn = s[j] * __expf(m[j] - Mn) + os * __expf(om - Mn);
            bool take = (om > m[j]) ||
                        (om == m[j] && oam >= 0 && (am[j] < 0 || oam < am[j]));
            am[j] = take ? oam : am[j];
            m[j] = Mn; s[j] = sn;
        }
    }

    if (lr == 0) {
#pragma unroll
        for (int j = 0; j < 8; ++j) {
            int r = j + half * 8;
            mbuf[w][r]  = m[j];
            sbuf[w][r]  = s[j];
            csbuf[w][r] = cs[j];
            pcbuf[w][r] = pc[j];
            ambuf[w][r] = am[j];
        }
    }
    __syncthreads();

    // Merge across the 8 waves (disjoint column subsets) and emit per-row.
    if (tid < 16) {
        int   r  = tid;
        float M  = NEG_INIT, S = 0.f, CS = 0.f, PC = 0.f;
        int   AM = -1;
        for (int ww = 0; ww < 8; ++ww) {
            float om  = mbuf[ww][r];
            float os  = sbuf[ww][r];
            int   oam = ambuf[ww][r];
            CS += csbuf[ww][r];
            PC += pcbuf[ww][r];
            float Mn = fmaxf(M, om);
            float Sn = S * __expf(M - Mn) + os * __expf(om - Mn);
            bool take = (om > M) || (om == M && oam >= 0 && (AM < 0 || oam < AM));
            AM = take ? oam : AM;
            M = Mn; S = Sn;
        }
        int gr    = rowBase + r;
        int label = gr ^ BB;
        float lse       = M + __logf(S);
        float pos_logit = PC * INV_T;
        loss_term[gr] = lse - pos_logit;
        correct[gr]   = (AM == label) ? 1.0f : 0.0f;
        cs_row[gr]    = CS;
        pc_row[gr]    = PC;
    }
}

// ---------------------------------------------------------------------------
// Kernel 4: deterministic final reductions -> 6 scalar outputs.
// ---------------------------------------------------------------------------
__global__ __launch_bounds__(256) void finalize_kernel(
    const float* __restrict__ loss_term, const float* __restrict__ correct,
    const float* __restrict__ cs_row, const float* __restrict__ pc_row,
    const double* __restrict__ colsum, const double* __restrict__ colsq,
    float* __restrict__ out)
{
    __shared__ double sh[6][256];
    const int t = threadIdx.x;
    double lsum = 0, csum = 0, cssum = 0, pcsum = 0;
    for (int r = t; r < NN; r += 256) {
        lsum  += (double)loss_term[r];
        csum  += (double)correct[r];
        cssum += (double)cs_row[r];
        pcsum += (double)pc_row[r];
    }
    double stdsum = 0, mnsum = 0;
    for (int d = t; d < DD; d += 256) {
        double sum = colsum[d], sq = colsq[d];
        double var = (sq - sum * sum / (double)NN) / (double)(NN - 1);
        stdsum += sqrt(var > 0.0 ? var : 0.0);
        double mean = sum / (double)NN;
        mnsum += mean * mean;
    }
    sh[0][t] = lsum; sh[1][t] = csum; sh[2][t] = cssum;
    sh[3][t] = pcsum; sh[4][t] = stdsum; sh[5][t] = mnsum;
    __syncthreads();
    for (int off = 128; off > 0; off >>= 1) {
        if (t < off)
            for (int q = 0; q < 6; ++q) sh[q][t] += sh[q][t + off];
        __syncthreads();
    }
    if (t == 0) {
        out[0] = (float)(sh[0][0] / (double)NN);                      // loss
        out[1] = (float)(sh[1][0] / (double)NN);                      // accuracy
        out[2] = (float)(sh[3][0] / (double)NN);                      // pos_sim
        out[3] = (float)(sh[2][0] / ((double)NN * (double)(NN - 1))); // neg_sim
        out[4] = (float)(sh[4][0] / (double)DD);                      // z_std
        out[5] = (float)sqrt(sh[5][0]);                               // z_mean_norm
    }
}

// ---------------------------------------------------------------------------
extern "C" void kernel_launch(void* const* d_in, const int* in_sizes, int n_in,
                              void* d_out, int out_size, void* d_ws, size_t ws_size,
                              hipStream_t stream) {
    (void)in_sizes; (void)n_in; (void)out_size; (void)ws_size;
    const float* za = (const float*)d_in[0];   // z_anchor  [4096,512] fp32
    const float* zp = (const float*)d_in[1];   // z_positive[4096,512] fp32
    float* out = (float*)d_out;                // 6 scalars

    char* ws = (char*)d_ws;
    float*  invnorm   = (float*)ws;                          // 8192 f  (32 KB)
    double* colsum    = (double*)(ws + 32768);               // 512 d   (4 KB)
    double* colsq     = (double*)(ws + 32768 + 4096);        // 512 d   (4 KB)
    float*  loss_term = (float*)(ws + 40960);                // 8192 f
    float*  correct   = loss_term + NN;
    float*  cs_row    = correct + NN;
    float*  pc_row    = cs_row + NN;

    row_invnorm_kernel<<<NN / 8, 256, 0, stream>>>(za, zp, invnorm);
    col_stats_kernel<<<DD, 256, 0, stream>>>(za, zp, colsum, colsq);
    ntxent_main_kernel<<<NN / 16, 256, 0, stream>>>(za, zp, invnorm,
                                                    loss_term, correct,
                                                    cs_row, pc_row);
    finalize_kernel<<<1, 256, 0, stream>>>(loss_term, correct, cs_row, pc_row,
                                           colsum, colsq, out);
}